// PyGConv_10977936409072
// MI455X (gfx1250) — compile-verified
//
#include <hip/hip_runtime.h>
#include <hip/hip_bf16.h>

typedef float v2f __attribute__((ext_vector_type(2)));
typedef float v8f __attribute__((ext_vector_type(8)));
typedef float v4f __attribute__((ext_vector_type(4)));

#define C128 128

// ---------------------------------------------------------------------------
// 1) out[n,c] = b[c]  (bias init; harness poisons d_out)   deg[i] = 1.0f
//    (self-loop contributes weight 1 to every node's degree)
// ---------------------------------------------------------------------------
__global__ void gcn_init(float* __restrict__ out, const float* __restrict__ b,
                         float* __restrict__ deg, int N) {
    long long t = (long long)blockIdx.x * blockDim.x + threadIdx.x;
    long long total = (long long)N * C128;
    if (t < total) out[t] = b[(int)(t & (C128 - 1))];
    if (t < N) deg[t] = 1.0f;
}

// ---------------------------------------------------------------------------
// 2) deg[dst[e]] += ew[e]
// ---------------------------------------------------------------------------
__global__ void gcn_degree(const int* __restrict__ dst, const float* __restrict__ ew,
                           float* __restrict__ deg, int E) {
    int e = blockIdx.x * blockDim.x + threadIdx.x;
    if (e < E) atomicAdd(&deg[dst[e]], ew[e]);
}

// ---------------------------------------------------------------------------
// 3) dis[i] = deg>0 ? rsqrt(deg) : 0   (in place)
// ---------------------------------------------------------------------------
__global__ void gcn_rsqrt(float* __restrict__ deg, int N) {
    int i = blockIdx.x * blockDim.x + threadIdx.x;
    if (i < N) {
        float d = deg[i];
        deg[i] = (d > 0.0f) ? rsqrtf(d) : 0.0f;
    }
}

// ---------------------------------------------------------------------------
// 4) norm[e] = dis[src]*ew*dis[dst]
// ---------------------------------------------------------------------------
__global__ void gcn_norm(const int* __restrict__ src, const int* __restrict__ dst,
                         const float* __restrict__ ew, const float* __restrict__ dis,
                         float* __restrict__ norm, int E) {
    int e = blockIdx.x * blockDim.x + threadIdx.x;
    if (e < E) norm[e] = dis[src[e]] * ew[e] * dis[dst[e]];
}

// ---------------------------------------------------------------------------
// 5) H = X (N x 128) @ W (128 x 128) with V_WMMA_F32_16X16X4_F32.
//    Block = 256 threads = 8 waves; block owns 16 rows, wave w owns cols
//    [16w, 16w+16). 32 k-steps of 4.
//    A frag: lanes 0-15 -> M = lane, K = {k0,k0+1}; lanes 16-31 -> K = {k0+2,k0+3}
//    B frag: lane&15 -> N-col, same K split.
//    D: VGPR r -> row r + 8*(lane>=16), col lane&15.
// ---------------------------------------------------------------------------
__global__ void gcn_gemm_wmma(const float* __restrict__ X, const float* __restrict__ W,
                              float* __restrict__ H, int N) {
    const int lane = threadIdx.x & 31;
    const int wave = threadIdx.x >> 5;
    const int half = lane >> 4;     // 0 or 1
    const int l15  = lane & 15;
    const int m0 = blockIdx.x * 16;
    const int n0 = wave * 16;

    int mA = m0 + l15;
    if (mA >= N) mA = N - 1;                 // clamp (N%16==0 for this problem)
    const float* xrow = X + (size_t)mA * C128 + half * 2;   // 8B-aligned
    const int n = n0 + l15;

    v8f acc = {};
#pragma unroll
    for (int k0 = 0; k0 < C128; k0 += 4) {
        v2f a = *(const v2f*)(xrow + k0);    // X[mA][k0+2*half .. +1]
        const int kb = k0 + half * 2;
        v2f bf;
        bf.x = W[kb * C128 + n];
        bf.y = W[(kb + 1) * C128 + n];
        acc = __builtin_amdgcn_wmma_f32_16x16x4_f32(
            /*neg_a=*/false, a, /*neg_b=*/false, bf,
            /*c_mod=*/(short)0, acc, /*reuse_a=*/false, /*reuse_b=*/false);
    }

#pragma unroll
    for (int r = 0; r < 8; ++r) {
        int m = m0 + r + half * 8;
        if (m < N) H[(size_t)m * C128 + n] = acc[r];
    }
}

// ---------------------------------------------------------------------------
// 6) Scatter: one 32-lane wave per message, 4 channels per lane (float4).
//    e < E:  out[dst[e], :] += h[src[e], :] * norm[e]
//    e >= E: self loop i=e-E: out[i, :] += h[i, :] * dis[i]^2
//    b128 gather (one 512B segment per wave) + 4 global_atomic_add_f32/lane,
//    all L2-resident (h: 51.2MB, out: 51.2MB, both < 192MB L2).
//    Index/norm loads are wave-uniform -> single L0 line broadcast.
// ---------------------------------------------------------------------------
__global__ void gcn_scatter(const float* __restrict__ h, const int* __restrict__ src,
                            const int* __restrict__ dst, const float* __restrict__ norm,
                            const float* __restrict__ dis, float* __restrict__ out,
                            int E, int N) {
    long long t = (long long)blockIdx.x * blockDim.x + threadIdx.x;
    int e = (int)(t >> 5);                 // message id
    int q = (int)(t & 31) * 4;             // channel base (0..124)
    if (e >= E + N) return;

    int s, d; float w;
    if (e < E) {
        s = src[e]; d = dst[e]; w = norm[e];
    } else {
        int i = e - E;
        s = i; d = i;
        float di = dis[i];
        w = di * di;                       // self-loop: dis*1*dis
    }

    v4f hv = *(const v4f*)(h + (size_t)s * C128 + q);   // 16B-aligned
    float* o = out + (size_t)d * C128 + q;
    atomicAdd(o + 0, hv.x * w);
    atomicAdd(o + 1, hv.y * w);
    atomicAdd(o + 2, hv.z * w);
    atomicAdd(o + 3, hv.w * w);
}

// ---------------------------------------------------------------------------
extern "C" void kernel_launch(void* const* d_in, const int* in_sizes, int n_in,
                              void* d_out, int out_size, void* d_ws, size_t ws_size,
                              hipStream_t stream) {
    const float* X  = (const float*)d_in[0];         // (1, N, 128)
    const int*   ei = (const int*)d_in[1];           // (2, E)
    const float* ew = (const float*)d_in[2];         // (E,)
    const float* W  = (const float*)d_in[3];         // (128, 128)
    const float* b  = (const float*)d_in[4];         // (128,)
    float* out = (float*)d_out;

    const int N = in_sizes[0] / C128;                // B=1
    const int E = in_sizes[2];
    const int* src = ei;
    const int* dst = ei + E;

    // workspace layout: H (N*128) | dis (N) | norm (E)
    float* H    = (float*)d_ws;
    float* dis  = H + (size_t)N * C128;
    float* norm = dis + N;

    const int T = 256;

    long long initTot = (long long)N * C128;
    gcn_init<<<(unsigned)((initTot + T - 1) / T), T, 0, stream>>>(out, b, dis, N);

    gcn_degree<<<(E + T - 1) / T, T, 0, stream>>>(dst, ew, dis, E);

    gcn_rsqrt<<<(N + T - 1) / T, T, 0, stream>>>(dis, N);

    gcn_norm<<<(E + T - 1) / T, T, 0, stream>>>(src, dst, ew, dis, norm, E);

    gcn_gemm_wmma<<<(N + 15) / 16, T, 0, stream>>>(X, W, H, N);

    long long scatTot = (long long)(E + N) * 32;     // 32 lanes per message
    gcn_scatter<<<(unsigned)((scatTot + T - 1) / T), T, 0, stream>>>(
        H, src, dst, norm, dis, out, E, N);
}